// RayDistributedSamplePoint_34076270527092
// MI455X (gfx1250) — compile-verified
//
#include <hip/hip_runtime.h>
#include <stdint.h>

#define FINE_NUM 128
#define N_COARSE 64
#define WPB 8                 // waves (= rays) per block
#define BLOCK (WPB * 32)

typedef __attribute__((ext_vector_type(4))) float f32x4;

// Match the async builtin's exact pointee type: int __attribute__((vector_size(16)))
typedef int v4i __attribute__((vector_size(4 * sizeof(int))));
typedef __attribute__((address_space(1))) v4i as1v4i;
typedef __attribute__((address_space(3))) v4i as3v4i;

#if defined(__gfx1250__) && __has_builtin(__builtin_amdgcn_global_load_async_to_lds_b128)
#define ASYNC_LDS 1
#else
#define ASYNC_LDS 0
#endif

__global__ __launch_bounds__(BLOCK) void nerf_fine_kernel(
    const float* __restrict__ rays,
    const float* __restrict__ depth,
    const float* __restrict__ density,
    float* __restrict__ out_t,
    float* __restrict__ out_p,
    int nrays)
{
  __shared__ float ldsZ[WPB * N_COARSE];   // coarse bins (depth) per ray
  __shared__ float ldsS[WPB * N_COARSE];   // density per ray
  __shared__ float ldsC[WPB * N_COARSE];   // cdf_s (64 entries, [0]=0) per ray

  const int lane = threadIdx.x & 31;
  // wave index is uniform per wave32; make that visible to the compiler so
  // `ray` lives in an SGPR (scalar guard, SMEM ray loads, SADDR-form stores).
  const int wv   = __builtin_amdgcn_readfirstlane((int)(threadIdx.x >> 5));
  const int ray  = blockIdx.x * WPB + wv;
  if (ray >= nrays) return;                // scalar (wave-uniform) exit
  const int wb   = wv * N_COARSE;

  // ---- Stage this ray's 64 depths + 64 densities (512 B) into LDS ----
#if ASYNC_LDS
  {
    // 32 lanes x 16B async copies: lanes 0-15 -> depth, lanes 16-31 -> density
    const float* gsrc = (lane < 16) ? (depth   + (size_t)ray * N_COARSE + lane * 4)
                                    : (density + (size_t)ray * N_COARSE + (lane - 16) * 4);
    float* ldst = (lane < 16) ? (ldsZ + wb + lane * 4)
                              : (ldsS + wb + (lane - 16) * 4);
    __builtin_amdgcn_global_load_async_to_lds_b128(
        (as1v4i*)(uintptr_t)(const void*)gsrc,
        (as3v4i*)(uint32_t)(uintptr_t)(void*)ldst,
        0, 0);
#if __has_builtin(__builtin_amdgcn_s_wait_asynccnt)
    __builtin_amdgcn_s_wait_asynccnt(0);
#else
    asm volatile("s_wait_asynccnt 0" ::: "memory");
#endif
  }
#else
  {
    const float2 zz = ((const float2*)(depth   + (size_t)ray * N_COARSE))[lane];
    const float2 ss = ((const float2*)(density + (size_t)ray * N_COARSE))[lane];
    ldsZ[wb + 2 * lane]     = zz.x;
    ldsZ[wb + 2 * lane + 1] = zz.y;
    ldsS[wb + 2 * lane]     = ss.x;
    ldsS[wb + 2 * lane + 1] = ss.y;
  }
#endif
  __builtin_amdgcn_wave_barrier();
  asm volatile("" ::: "memory");           // keep LDS reads after staging

  // ---- Weights: lane owns coarse samples k0=2*lane, k1=2*lane+1 ----
  const int k0 = lane * 2, k1 = k0 + 1;
  const float z0 = ldsZ[wb + k0];
  const float z1 = ldsZ[wb + k1];
  const float s0 = ldsS[wb + k0];
  const float s1 = ldsS[wb + k1];
  const float zn = (lane < 31) ? ldsZ[wb + k1 + 1] : 0.0f;

  const float d0 = z1 - z0;
  const float d1 = (lane < 31) ? (zn - z1) : 1.0e10f;

  const float t0 = fmaxf(s0, 0.0f) * d0;
  const float t1 = fmaxf(s1, 0.0f) * d1;
  const float e0 = __expf(-t0);            // = 1 - alpha
  const float e1 = __expf(-t1);
  const float a0 = 1.0f - e0;
  const float a1 = 1.0f - e1;
  const float lg0 = __logf(e0 + 1.0e-10f); // log(1 - alpha + 1e-10)
  const float lg1 = __logf(e1 + 1.0e-10f);

  // cumprod via log-space exclusive wave scan (Hillis-Steele, wave32)
  const float pairA = lg0 + lg1;
  float scanA = pairA;
#pragma unroll
  for (int off = 1; off < 32; off <<= 1) {
    float t = __shfl_up(scanA, off, 32);
    if (lane >= off) scanA += t;
  }
  const float exA = scanA - pairA;         // sum of logs over samples < k0
  const float tr0 = __expf(exA);
  const float tr1 = __expf(exA + lg0);

  const float w0 = a0 * tr0 + 1.0e-5f;
  const float w1 = a1 * tr1 + 1.0e-5f;

  // CDF over weights[1:]: zero out sample 0, inclusive wave scan, normalize
  const float u0 = (lane == 0) ? 0.0f : w0;
  const float u1 = w1;
  const float pairB = u0 + u1;
  float scanB = pairB;
#pragma unroll
  for (int off = 1; off < 32; off <<= 1) {
    float t = __shfl_up(scanB, off, 32);
    if (lane >= off) scanB += t;
  }
  const float exB   = scanB - pairB;
  const float total = __shfl(scanB, 31, 32);
  const float invS  = __builtin_amdgcn_rcpf(total);   // v_rcp_f32
  const float ctop  = total * invS;                   // == stored cdf_s[63]

  ldsC[wb + k0] = (exB + u0) * invS;       // cdf_s[k0]  (lane0,k0 -> 0 exactly)
  ldsC[wb + k1] = scanB * invS;            // cdf_s[k1]
  __builtin_amdgcn_wave_barrier();
  asm volatile("" ::: "memory");

  // ---- Inverse-CDF sample 128 fine bins: 4 per lane ----
  const float* rp = rays + (size_t)ray * 6;            // uniform -> SMEM
  const float ox = rp[0], oy = rp[1], ozr = rp[2];
  const float dx = rp[3], dy = rp[4], dzr = rp[5];

  float ft[4];
#pragma unroll
  for (int j = 0; j < 4; ++j) {
    const int   i = lane * 4 + j;
    const float f = (float)i * (1.0f / 127.0f);
    // searchsorted(cdf_s, f, side='left') over 64 entries: 6-step binary search
    int lo = 0, hi = N_COARSE;
#pragma unroll
    for (int s = 0; s < 6; ++s) {
      const int mid = (lo + hi) >> 1;
      const float cm = ldsC[wb + mid];
      if (cm < f) lo = mid + 1; else hi = mid;
    }
    const bool nm    = (f <= 0.0f) || (f > ctop);
    const int  above = nm ? 1 : lo;        // nm covers the lo==64 case (f>ctop)
    const int  below = above - 1;          // above >= 1 always (cdf_s[0]==0 < f)
    const float cb = ldsC[wb + below];
    const float ca = ldsC[wb + above];
    const float sb = ldsZ[wb + below];
    const float sa = ldsZ[wb + above];
    float dn = ca - cb;
    dn = (dn < 1.0e-5f) ? 1.0f : dn;
    const float dd = (f - cb) * __builtin_amdgcn_rcpf(dn);
    ft[j] = (sa - sb) * dd + sb;
  }

  // ---- Streaming (nontemporal) coalesced stores ----
  f32x4 tvec = { ft[0], ft[1], ft[2], ft[3] };
  __builtin_nontemporal_store(tvec, (f32x4*)(out_t + (size_t)ray * FINE_NUM + lane * 4));

  float p[12];
#pragma unroll
  for (int j = 0; j < 4; ++j) {
    p[3 * j + 0] = fmaf(ft[j], dx,  ox);
    p[3 * j + 1] = fmaf(ft[j], dy,  oy);
    p[3 * j + 2] = fmaf(ft[j], dzr, ozr);
  }
  float* pb = out_p + (size_t)ray * (FINE_NUM * 3) + lane * 12;  // 48B/lane, 16B aligned
  f32x4 v0 = { p[0], p[1], p[2],  p[3]  };
  f32x4 v1 = { p[4], p[5], p[6],  p[7]  };
  f32x4 v2 = { p[8], p[9], p[10], p[11] };
  __builtin_nontemporal_store(v0, (f32x4*)(pb + 0));
  __builtin_nontemporal_store(v1, (f32x4*)(pb + 4));
  __builtin_nontemporal_store(v2, (f32x4*)(pb + 8));
}

extern "C" void kernel_launch(void* const* d_in, const int* in_sizes, int n_in,
                              void* d_out, int out_size, void* d_ws, size_t ws_size,
                              hipStream_t stream) {
  (void)n_in; (void)out_size; (void)d_ws; (void)ws_size;
  const float* rays    = (const float*)d_in[0];
  const float* depth   = (const float*)d_in[1];
  const float* density = (const float*)d_in[2];
  const int nrays = in_sizes[0] / 6;               // 131072
  float* out_t = (float*)d_out;                    // (N, 128)
  float* out_p = out_t + (size_t)nrays * FINE_NUM; // (N, 128, 3)
  const int blocks = (nrays + WPB - 1) / WPB;      // 16384 blocks of 8 waves
  nerf_fine_kernel<<<blocks, BLOCK, 0, stream>>>(rays, depth, density, out_t, out_p, nrays);
}